// Net_72782515798161
// MI455X (gfx1250) — compile-verified
//
#include <hip/hip_runtime.h>
#include <hip/hip_bf16.h>

// ---------------------------------------------------------------------------
// Graph U-Net forward for MI455X (gfx1250).
//   GEMMs        -> V_WMMA_F32_16X16X4_F32 (full fp32 matrix pipe), A tile
//                   staged into LDS via the Tensor Data Mover (TENSORcnt).
//   Aggregation  -> GLOBAL_ATOMIC_ADD_F32 scatter (bandwidth bound)
//   TopK         -> bitonic sort of composite 64-bit keys (stable, matches
//                   jax.lax.top_k ordering); j<=256 stages fused in LDS.
// ---------------------------------------------------------------------------

typedef __attribute__((ext_vector_type(2))) float v2f;
typedef __attribute__((ext_vector_type(8))) float v8f;
typedef unsigned int u32x4 __attribute__((ext_vector_type(4)));
typedef int          i32x8 __attribute__((ext_vector_type(8)));
typedef int          i32x4 __attribute__((ext_vector_type(4)));

#define NNODES 100000
#define NEDGES 1600000
#define EP     (NEDGES + NNODES)   // edges after self-loop append = 1,700,000
#define FDIM   64
#define HDIM   128
#define K1     50000
#define K2     25000
#define K3     12500

static inline int nblk(long long total, int bs) { return (int)((total + bs - 1) / bs); }

// ---------------------------------------------------------------- utilities
__global__ void k_zero_f(float* p, long long n) {
    long long t = (long long)blockIdx.x * blockDim.x + threadIdx.x;
    if (t < n) p[t] = 0.0f;
}
__global__ void k_zero_i(int* p, long long n) {
    long long t = (long long)blockIdx.x * blockDim.x + threadIdx.x;
    if (t < n) p[t] = 0;
}
__global__ void k_copy_f(float* dst, const float* src, long long n) {
    long long t = (long long)blockIdx.x * blockDim.x + threadIdx.x;
    if (t < n) dst[t] = src[t];
}
__global__ void k_copy_i(int* dst, const int* src, long long n) {
    long long t = (long long)blockIdx.x * blockDim.x + threadIdx.x;
    if (t < n) dst[t] = src[t];
}

// ------------------------------------------------ edge list with self loops
__global__ void k_build_edges(const int* esrc, const int* edst,
                              int* src, int* dst, float* ew) {
    int i = blockIdx.x * blockDim.x + threadIdx.x;
    if (i >= EP) return;
    if (i < NEDGES) {
        int s = esrc[i], d = edst[i];
        src[i] = s; dst[i] = d;
        ew[i] = (s != d) ? 1.0f : 0.0f;      // remove_self_loops
    } else {
        int v = i - NEDGES;                   // add_self_loops
        src[i] = v; dst[i] = v; ew[i] = 1.0f;
    }
}

// ---------------------------------------------------------- GCN components
__global__ void k_degree(const int* dst, const float* ew, float* deg, int ep) {
    int e = blockIdx.x * blockDim.x + threadIdx.x;
    if (e < ep) unsafeAtomicAdd(&deg[dst[e]], ew[e]);
}
__global__ void k_dinv(const float* deg, float* dinv, int n) {
    int i = blockIdx.x * blockDim.x + threadIdx.x;
    if (i < n) {
        float d = deg[i];
        dinv[i] = (d > 0.0f) ? rsqrtf(d) : 0.0f;
    }
}

// fp32 WMMA GEMM: C[M,N] = A[M,K] @ B[K,N], row-major, K<=128, N%16==0.
// Block = (N/16) waves over one 16-row stripe. The shared 16xK A tile is
// DMA'd into LDS once per block by the Tensor Data Mover (wave 0 issues it;
// TDM ignores EXEC so exactly one wave must execute the instruction), with
// tensor_dim1 = M-m0 so out-of-range tail rows read as zero.
__global__ void k_gemm_f32_wmma(const float* __restrict__ A,
                                const float* __restrict__ B,
                                float* __restrict__ C, int M, int K, int N) {
    __shared__ __align__(16) float sa[16 * 128];
    const int wave = threadIdx.x >> 5;
    const int lane = threadIdx.x & 31;
    const int half = lane >> 4;          // 0: K=0,1   1: K=2,3
    const int l    = lane & 15;
    const int m0   = blockIdx.x * 16;

#if __has_builtin(__builtin_amdgcn_tensor_load_to_lds) && \
    __has_builtin(__builtin_amdgcn_s_wait_tensorcnt)
    if (threadIdx.x < 32) {
        const unsigned long long ga =
            (unsigned long long)(const void*)(A + (unsigned long long)m0 * K);
        const unsigned lds_b = (unsigned)(unsigned long long)(const void*)sa;
        const unsigned dim1  = (unsigned)(M - m0);   // rows left -> OOB reads 0
        // D# group0: count=1 | lds_addr | global_addr[56:0] | type=2
        u32x4 g0 = { 1u, lds_b, (unsigned)(ga & 0xffffffffull),
                     (unsigned)((ga >> 32) & 0x1ffffffull) | (2u << 30) };
        // D# group1: data_size=4B; tensor_dim0=K; tensor_dim1=dim1;
        //            tile_dim0=K; tile_dim1=16; tensor_dim0_stride=K
        i32x8 g1 = { (int)0x00020000,
                     (int)(((unsigned)K & 0xffffu) << 16),
                     (int)(((unsigned)K >> 16) | ((dim1 & 0xffffu) << 16)),
                     (int)((dim1 >> 16) | (((unsigned)K & 0xffffu) << 16)),
                     16,
                     K,
                     0, 0 };
        i32x4 gz4 = { 0, 0, 0, 0 };
        i32x8 gz8 = { 0, 0, 0, 0, 0, 0, 0, 0 };
        // clang-23 / therock-10.0 lane: 6-arg form
        __builtin_amdgcn_tensor_load_to_lds(g0, g1, gz4, gz4, gz8, 0);
        __builtin_amdgcn_s_wait_tensorcnt(0);
    }
    __syncthreads();
#else
    for (int t = threadIdx.x; t < 16 * K; t += blockDim.x) {
        int r = t / K, c = t - r * K;
        int row = m0 + r;
        sa[t] = (row < M) ? A[(long long)row * K + c] : 0.0f;
    }
    __syncthreads();
#endif

    const int n0   = wave * 16;
    const int colB = n0 + l;
    v8f acc = {};
    for (int kk = 0; kk < K; kk += 4) {
        const int ka = kk + half * 2;
        v2f a, b;
        a.x = sa[l * K + ka];
        a.y = sa[l * K + ka + 1];
        b.x = B[(long long)ka * N + colB];
        b.y = B[(long long)(ka + 1) * N + colB];
        acc = __builtin_amdgcn_wmma_f32_16x16x4_f32(
            false, a, false, b, (short)0, acc, false, false);
    }
#pragma unroll
    for (int r = 0; r < 8; ++r) {
        int row = m0 + r + half * 8;
        if (row < M) C[(long long)row * N + n0 + l] = acc[r];
    }
}

// out[dst] += h[src] * norm  -- one wave per edge, lanes stride features.
__global__ void k_scatter(const int* __restrict__ src, const int* __restrict__ dst,
                          const float* __restrict__ ew, const float* __restrict__ dinv,
                          const float* __restrict__ h, float* __restrict__ out,
                          int ep, int hd) {
    long long t = (long long)blockIdx.x * blockDim.x + threadIdx.x;
    int lane = (int)(t & 31);
    long long e = t >> 5;
    if (e >= ep) return;
    float w = ew[e];
    if (w == 0.0f) return;
    int s = src[e], d = dst[e];
    float nrm = dinv[s] * w * dinv[d];
    if (nrm == 0.0f) return;
    const float* hs = h + (long long)s * hd;
    float* od = out + (long long)d * hd;
    for (int f = lane; f < hd; f += 32)
        unsafeAtomicAdd(&od[f], hs[f] * nrm);
}

__global__ void k_bias_act(float* y, const float* b, long long total, int od, int relu) {
    long long t = (long long)blockIdx.x * blockDim.x + threadIdx.x;
    if (t >= total) return;
    float v = y[t] + b[t % od];
    if (relu) v = fmaxf(v, 0.0f);
    y[t] = v;
}

// ---------------------------------------------------------------- pooling
__global__ void k_wnorm(const float* w, float* inv) {
    __shared__ float s[128];
    float v = w[threadIdx.x];
    s[threadIdx.x] = v * v;
    __syncthreads();
    for (int o = 64; o > 0; o >>= 1) {
        if ((int)threadIdx.x < o) s[threadIdx.x] += s[threadIdx.x + o];
        __syncthreads();
    }
    if (threadIdx.x == 0) inv[0] = 1.0f / sqrtf(s[0]);
}

__global__ void k_score(const float* __restrict__ x, const float* __restrict__ w,
                        const float* __restrict__ invn, float* __restrict__ score, int n) {
    long long t = (long long)blockIdx.x * blockDim.x + threadIdx.x;
    int lane = (int)(t & 31);
    long long i = t >> 5;
    if (i >= n) return;
    float p = 0.0f;
    for (int f = lane; f < HDIM; f += 32)
        p += x[i * HDIM + f] * w[f];
    for (int off = 16; off > 0; off >>= 1)
        p += __shfl_down(p, off, 32);
    if (lane == 0) score[i] = tanhf(p * invn[0]);
}

// composite key: (~monotone(score)) << 32 | index   (ascending sort =>
// score descending, index ascending on ties -- matches jax.lax.top_k)
__global__ void k_make_keys(const float* score, unsigned long long* keys,
                            int n, int npad) {
    int i = blockIdx.x * blockDim.x + threadIdx.x;
    if (i >= npad) return;
    if (i >= n) { keys[i] = ~0ULL; return; }
    unsigned u = __float_as_uint(score[i]);
    u = (u & 0x80000000u) ? ~u : (u | 0x80000000u);   // monotone ascending map
    keys[i] = ((unsigned long long)(~u) << 32) | (unsigned)i;
}

__global__ void k_bitonic(unsigned long long* keys, int npad, int k, int j) {
    int i = blockIdx.x * blockDim.x + threadIdx.x;
    if (i >= npad) return;
    int p = i ^ j;
    if (p > i) {
        unsigned long long a = keys[i], b = keys[p];
        bool asc = ((i & k) == 0);
        if ((asc && a > b) || (!asc && a < b)) { keys[i] = b; keys[p] = a; }
    }
}

// fused bitonic sub-stages j0..1 (j0 <= 256) on a 512-element LDS span
__global__ void k_bitonic_shared(unsigned long long* keys, int k, int j0) {
    __shared__ unsigned long long sk[512];
    const int base = blockIdx.x * 512;
    sk[threadIdx.x]       = keys[base + threadIdx.x];
    sk[threadIdx.x + 256] = keys[base + threadIdx.x + 256];
    __syncthreads();
    for (int j = j0; j > 0; j >>= 1) {
        for (int e = threadIdx.x; e < 512; e += 256) {
            int p = e ^ j;
            if (p > e) {
                bool asc = (((base + e) & k) == 0);
                unsigned long long a = sk[e], b = sk[p];
                if ((asc && a > b) || (!asc && a < b)) { sk[e] = b; sk[p] = a; }
            }
        }
        __syncthreads();
    }
    keys[base + threadIdx.x]       = sk[threadIdx.x];
    keys[base + threadIdx.x + 256] = sk[threadIdx.x + 256];
}

__global__ void k_pool_apply(const unsigned long long* __restrict__ keys,
                             const float* __restrict__ score,
                             const float* __restrict__ xin, float* __restrict__ xout,
                             int* __restrict__ perm, int* __restrict__ keep,
                             int* __restrict__ nid, int kc) {
    long long t = (long long)blockIdx.x * blockDim.x + threadIdx.x;
    if (t >= (long long)kc * HDIM) return;
    int r = (int)(t / HDIM), f = (int)(t % HDIM);
    int idx = (int)(unsigned)(keys[r] & 0xffffffffULL);
    xout[(long long)r * HDIM + f] = xin[(long long)idx * HDIM + f] * score[idx];
    if (f == 0) { perm[r] = idx; keep[idx] = 1; nid[idx] = r; }
}

__global__ void k_remap_edges(const int* src, const int* dst, const float* ew,
                              const int* keep, const int* nid,
                              int* nsrc, int* ndst, float* new_ew) {
    int e = blockIdx.x * blockDim.x + threadIdx.x;
    if (e >= EP) return;
    int s = src[e], d = dst[e];
    int valid = keep[s] & keep[d];
    nsrc[e] = nid[s];            // nid==0 for dropped nodes (zero-init)
    ndst[e] = nid[d];
    new_ew[e] = ew[e] * (float)valid;
}

// unpool: xbig (already = res) ; xbig[perm[r]] += xsmall[r]  (perm unique)
__global__ void k_unpool_scatter(const int* perm, const float* xs, float* xbig, int kc) {
    long long t = (long long)blockIdx.x * blockDim.x + threadIdx.x;
    if (t >= (long long)kc * HDIM) return;
    int r = (int)(t / HDIM), f = (int)(t % HDIM);
    xbig[(long long)perm[r] * HDIM + f] += xs[(long long)r * HDIM + f];
}

// ---------------------------------------------------------------------------
// Host orchestration (all on `stream`, graph-capture safe).
// ---------------------------------------------------------------------------
static void run_gcn(const float* x, int n, int K, int Nout,
                    const float* W, const float* b,
                    const int* src, const int* dst, const float* ew,
                    float* hbuf, float* deg, float* dinv, float* out,
                    int relu, hipStream_t st) {
    k_gemm_f32_wmma<<<dim3(nblk(n, 16)), dim3(32 * (Nout / 16)), 0, st>>>(x, W, hbuf, n, K, Nout);
    k_zero_f<<<nblk(n, 256), 256, 0, st>>>(deg, n);
    k_degree<<<nblk(EP, 256), 256, 0, st>>>(dst, ew, deg, EP);
    k_dinv<<<nblk(n, 256), 256, 0, st>>>(deg, dinv, n);
    long long tot = (long long)n * Nout;
    k_zero_f<<<nblk(tot, 256), 256, 0, st>>>(out, tot);
    k_scatter<<<nblk((long long)EP * 32, 256), 256, 0, st>>>(src, dst, ew, dinv, hbuf, out, EP, Nout);
    k_bias_act<<<nblk(tot, 256), 256, 0, st>>>(out, b, tot, Nout, relu);
}

static int next_pow2(int n) { int p = 1; while (p < n) p <<= 1; return p; }

static void run_pool(const float* xin, int n, int kc, const float* pw,
                     const int* src, const int* dst, const float* ew,
                     int* nsrc, int* ndst, float* new_ew,
                     float* xout, int* perm,
                     float* score, unsigned long long* keys, int* keep, int* nid,
                     float* invw, hipStream_t st) {
    k_wnorm<<<1, 128, 0, st>>>(pw, invw);
    k_score<<<nblk((long long)n * 32, 256), 256, 0, st>>>(xin, pw, invw, score, n);
    int npad = next_pow2(n);                  // >= 512 for all levels here
    k_make_keys<<<nblk(npad, 256), 256, 0, st>>>(score, keys, n, npad);
    for (int k = 2; k <= npad; k <<= 1) {
        for (int j = k >> 1; j >= 512; j >>= 1)
            k_bitonic<<<nblk(npad, 256), 256, 0, st>>>(keys, npad, k, j);
        int j0 = (k >> 1 < 256) ? (k >> 1) : 256;
        k_bitonic_shared<<<npad / 512, 256, 0, st>>>(keys, k, j0);
    }
    k_zero_i<<<nblk(n, 256), 256, 0, st>>>(keep, n);
    k_zero_i<<<nblk(n, 256), 256, 0, st>>>(nid, n);
    k_pool_apply<<<nblk((long long)kc * HDIM, 256), 256, 0, st>>>(keys, score, xin, xout,
                                                                  perm, keep, nid, kc);
    k_remap_edges<<<nblk(EP, 256), 256, 0, st>>>(src, dst, ew, keep, nid, nsrc, ndst, new_ew);
}

extern "C" void kernel_launch(void* const* d_in, const int* in_sizes, int n_in,
                              void* d_out, int out_size, void* d_ws, size_t ws_size,
                              hipStream_t stream) {
    (void)in_sizes; (void)n_in; (void)out_size; (void)ws_size;
    const float* x_in  = (const float*)d_in[0];
    const int*   ei    = (const int*)d_in[1];         // (2, E)
    const float* dW0 = (const float*)d_in[3],  *db0 = (const float*)d_in[4];
    const float* dW1 = (const float*)d_in[5],  *db1 = (const float*)d_in[6];
    const float* dW2 = (const float*)d_in[7],  *db2 = (const float*)d_in[8];
    const float* dW3 = (const float*)d_in[9],  *db3 = (const float*)d_in[10];
    const float* pw1 = (const float*)d_in[11], *pw2 = (const float*)d_in[12],
               * pw3 = (const float*)d_in[13];
    const float* uW1 = (const float*)d_in[14], *ub1 = (const float*)d_in[15];
    const float* uW2 = (const float*)d_in[16], *ub2 = (const float*)d_in[17];
    const float* uW3 = (const float*)d_in[18], *ub3 = (const float*)d_in[19];

    // ---- workspace bump allocator -----------------------------------------
    size_t off = 0;
    auto alloc = [&](size_t bytes) -> void* {
        off = (off + 255) & ~(size_t)255;
        void* p = (char*)d_ws + off;
        off += bytes;
        return p;
    };
    int   *s0 = (int*)alloc(EP * 4), *d0 = (int*)alloc(EP * 4);
    int   *s1 = (int*)alloc(EP * 4), *d1 = (int*)alloc(EP * 4);
    int   *s2 = (int*)alloc(EP * 4), *d2 = (int*)alloc(EP * 4);
    int   *s3 = (int*)alloc(EP * 4), *d3 = (int*)alloc(EP * 4);
    float *w0 = (float*)alloc(EP * 4), *w1 = (float*)alloc(EP * 4);
    float *w2 = (float*)alloc(EP * 4), *w3 = (float*)alloc(EP * 4);
    float *hbuf = (float*)alloc((size_t)NNODES * HDIM * 4);
    float *xA   = (float*)alloc((size_t)NNODES * HDIM * 4);
    float *xB   = (float*)alloc((size_t)NNODES * HDIM * 4);
    float *xs0  = (float*)alloc((size_t)NNODES * HDIM * 4);
    float *xs1  = (float*)alloc((size_t)K1 * HDIM * 4);
    float *xs2  = (float*)alloc((size_t)K2 * HDIM * 4);
    float *xlat = (float*)alloc((size_t)K3 * HDIM * 4);
    float *deg  = (float*)alloc(NNODES * 4);
    float *dinv = (float*)alloc(NNODES * 4);
    float *score = (float*)alloc(NNODES * 4);
    float *invw  = (float*)alloc(256);
    unsigned long long *keys = (unsigned long long*)alloc((size_t)131072 * 8);
    int *perm1 = (int*)alloc(K1 * 4), *perm2 = (int*)alloc(K2 * 4),
        *perm3 = (int*)alloc(K3 * 4);
    int *keep = (int*)alloc(NNODES * 4), *nid = (int*)alloc(NNODES * 4);

    // ---- encoder ----------------------------------------------------------
    k_build_edges<<<nblk(EP, 256), 256, 0, stream>>>(ei, ei + NEDGES, s0, d0, w0);
    run_gcn(x_in, NNODES, FDIM, HDIM, dW0, db0, s0, d0, w0, hbuf, deg, dinv, xs0, 1, stream);

    run_pool(xs0, NNODES, K1, pw1, s0, d0, w0, s1, d1, w1, xA, perm1,
             score, keys, keep, nid, invw, stream);
    run_gcn(xA, K1, HDIM, HDIM, dW1, db1, s1, d1, w1, hbuf, deg, dinv, xs1, 1, stream);

    run_pool(xs1, K1, K2, pw2, s1, d1, w1, s2, d2, w2, xA, perm2,
             score, keys, keep, nid, invw, stream);
    run_gcn(xA, K2, HDIM, HDIM, dW2, db2, s2, d2, w2, hbuf, deg, dinv, xs2, 1, stream);

    run_pool(xs2, K2, K3, pw3, s2, d2, w2, s3, d3, w3, xA, perm3,
             score, keys, keep, nid, invw, stream);
    run_gcn(xA, K3, HDIM, HDIM, dW3, db3, s3, d3, w3, hbuf, deg, dinv, xlat, 1, stream);

    // ---- decoder ----------------------------------------------------------
    // up0: res = xs2 (25000)
    k_copy_f<<<nblk((long long)K2 * HDIM, 256), 256, 0, stream>>>(xB, xs2, (long long)K2 * HDIM);
    k_unpool_scatter<<<nblk((long long)K3 * HDIM, 256), 256, 0, stream>>>(perm3, xlat, xB, K3);
    run_gcn(xB, K2, HDIM, HDIM, uW1, ub1, s2, d2, w2, hbuf, deg, dinv, xA, 1, stream);

    // up1: res = xs1 (50000)
    k_copy_f<<<nblk((long long)K1 * HDIM, 256), 256, 0, stream>>>(xB, xs1, (long long)K1 * HDIM);
    k_unpool_scatter<<<nblk((long long)K2 * HDIM, 256), 256, 0, stream>>>(perm2, xA, xB, K2);
    run_gcn(xB, K1, HDIM, HDIM, uW2, ub2, s1, d1, w1, hbuf, deg, dinv, xA, 1, stream);

    // up2: res = xs0 (100000), output dim F=64, no relu
    k_copy_f<<<nblk((long long)NNODES * HDIM, 256), 256, 0, stream>>>(xB, xs0, (long long)NNODES * HDIM);
    k_unpool_scatter<<<nblk((long long)K1 * HDIM, 256), 256, 0, stream>>>(perm1, xA, xB, K1);
    run_gcn(xB, NNODES, HDIM, FDIM, uW3, ub3, s0, d0, w0, hbuf, deg, dinv, xA, 0, stream);

    // ---- pack outputs: (x, latent_x, latent_edge, latent_batch) -----------
    float* outf = (float*)d_out;
    int*   outi = (int*)d_out;
    long long o0 = 0;
    long long o1 = o0 + (long long)NNODES * FDIM;      // 6,400,000
    long long o2 = o1 + (long long)K3 * HDIM;          // 8,000,000
    long long o3 = o2 + (long long)EP;                 // 9,700,000
    long long o4 = o3 + (long long)EP;                 // 11,400,000
    k_copy_f<<<nblk(o1 - o0, 256), 256, 0, stream>>>(outf + o0, xA, o1 - o0);
    k_copy_f<<<nblk(o2 - o1, 256), 256, 0, stream>>>(outf + o1, xlat, o2 - o1);
    k_copy_i<<<nblk(EP, 256), 256, 0, stream>>>(outi + o2, s3, EP);
    k_copy_i<<<nblk(EP, 256), 256, 0, stream>>>(outi + o3, d3, EP);
    k_zero_i<<<nblk(K3, 256), 256, 0, stream>>>(outi + o4, K3);
}